// MyModel_87522843559978
// MI455X (gfx1250) — compile-verified
//
#include <hip/hip_runtime.h>
#include <hip/hip_bf16.h>
#include <math.h>

typedef __attribute__((ext_vector_type(2))) float v2f;
typedef __attribute__((ext_vector_type(8))) float v8f;

#define KTAGS 5

// ---------------------------------------------------------------------------
// Fused forward kernel. Blocks [0, gv) run the Viterbi (max-plus) recursion,
// blocks [gv, gv+gf) run the probability-domain forward recursion via WMMA.
// Both stream the same potentials array concurrently -> L2 reuse (80MB fits
// in the 192MB L2), and 768 concurrent waves instead of two serialized
// underfilled launches.
// ---------------------------------------------------------------------------
__global__ void crf_fwd_fused(const float* __restrict__ pot,
                              const float* __restrict__ trans,
                              const int* __restrict__ lens,
                              unsigned short* __restrict__ bp_out, // [(T-1)][B]
                              int* __restrict__ last_tag,
                              float* __restrict__ lognorm,
                              int B, int T, int gv)
{
    if ((int)blockIdx.x < gv) {
        // ----------------- Viterbi forward: one thread per batch -----------
        int b = blockIdx.x * blockDim.x + threadIdx.x;
        if (b >= B) return;

        float tr[KTAGS * KTAGS];
#pragma unroll
        for (int i = 0; i < KTAGS * KTAGS; ++i) tr[i] = trans[i];

        const float* pb = pot + (size_t)b * T * KTAGS;
        float al[KTAGS];
#pragma unroll
        for (int j = 0; j < KTAGS; ++j) al[j] = pb[j];
        int len = lens[b];

        for (int t = 1; t < T; ++t) {
            const float* pt = pb + (size_t)t * KTAGS;
            int tpf = (t + 8 < T) ? (t + 8) : (T - 1);
            __builtin_prefetch(pb + (size_t)tpf * KTAGS, 0, 3); // WGP-scope

            float p[KTAGS];
#pragma unroll
            for (int j = 0; j < KTAGS; ++j) p[j] = pt[j];

            bool valid = t < len;
            unsigned bits = 0;
            float na[KTAGS];
#pragma unroll
            for (int j = 0; j < KTAGS; ++j) {
                float best = al[0] + tr[0 * KTAGS + j];
                int bi = 0;
#pragma unroll
                for (int i = 1; i < KTAGS; ++i) {
                    float s = al[i] + tr[i * KTAGS + j];
                    if (s > best) { best = s; bi = i; } // strict '>' = first argmax
                }
                na[j] = best + p[j];
                int bpj = valid ? bi : j;               // identity bp when invalid
                bits |= (unsigned)bpj << (3 * j);
            }
#pragma unroll
            for (int j = 0; j < KTAGS; ++j) al[j] = valid ? na[j] : al[j];
            bp_out[(size_t)(t - 1) * B + b] = (unsigned short)bits;
        }

        float best = al[0]; int bi = 0;
#pragma unroll
        for (int j = 1; j < KTAGS; ++j) if (al[j] > best) { best = al[j]; bi = j; }
        last_tag[b] = bi;
        return;
    }

    // ----------------- Forward algorithm via WMMA: 16 batches / wave -------
    int tid  = (blockIdx.x - gv) * blockDim.x + threadIdx.x;
    int wave = tid >> 5;
    int lane = threadIdx.x & 31;
    int half = lane >> 4;        // 0: lanes 0-15, 1: lanes 16-31
    int j    = lane & 15;        // tag row for A / batch column for B,D
    int bb   = wave * 16 + j;
    int bbc  = (bb < B) ? bb : (B - 1);

    // A = exp(trans)^T in 16x4 A-layout. Rows j>=5 zero so D rows 5..15 = 0.
    float a0 = 0.f, a1 = 0.f, a2 = 0.f, a3 = 0.f;
    if (j < KTAGS) {
        int k0 = half * 2;                                // K = {0,1} / {2,3}
        a0 = __expf(trans[k0 * KTAGS + j]);
        a1 = __expf(trans[(k0 + 1) * KTAGS + j]);
        if (!half) a2 = __expf(trans[4 * KTAGS + j]);     // K=4 (K=5..7 zero)
    }

    const float* pb = pot + (size_t)bbc * T * KTAGS;
    int len = lens[bbc];

    // t = 0 init: alpha = pot[:,0]; scale so max prob == 1.
    float q0 = pb[0], q1 = pb[1], q2 = pb[2], q3 = pb[3], q4 = pb[4];
    float c  = fmaxf(fmaxf(fmaxf(q0, q1), fmaxf(q2, q3)), q4);
    float p0 = __expf(q0 - c), p1 = __expf(q1 - c), p2 = __expf(q2 - c),
          p3 = __expf(q3 - c), p4 = __expf(q4 - c);
    float logacc = c;

    for (int t = 1; t < T; ++t) {
        // Build B (4x16, rows K across lanes, D-style half split: VGPR0 holds
        // rows {0,2}, VGPR1 rows {1,3}; second WMMA holds rows {4,6}/{5,7}).
        float p2x = __shfl_xor(p2, 16, 32);
        float p3x = __shfl_xor(p3, 16, 32);
        v2f Blo, Bhi, Alo, Ahi;
        Blo[0] = half ? p2x : p0;  Blo[1] = half ? p3x : p1;
        Bhi[0] = half ? 0.f : p4;  Bhi[1] = 0.f;
        Alo[0] = a0; Alo[1] = a1;
        Ahi[0] = a2; Ahi[1] = a3;

        v8f acc = {};
        acc = __builtin_amdgcn_wmma_f32_16x16x4_f32(false, Alo, false, Blo,
                                                    (short)0, acc, false, false);
        acc = __builtin_amdgcn_wmma_f32_16x16x4_f32(false, Ahi, false, Bhi,
                                                    (short)0, acc, false, false);
        // D row j (tag), col = batch lane: q_j = acc[j] at lanes 0..15.

        const float* pt = pb + (size_t)t * KTAGS;
        int tpf = (t + 8 < T) ? (t + 8) : (T - 1);
        __builtin_prefetch(pb + (size_t)tpf * KTAGS, 0, 3);

        float w0 = acc[0] * __expf(pt[0]);
        float w1 = acc[1] * __expf(pt[1]);
        float w2 = acc[2] * __expf(pt[2]);
        float w3 = acc[3] * __expf(pt[3]);
        float w4 = acc[4] * __expf(pt[4]);
        float m  = fmaxf(fmaxf(fmaxf(w0, w1), fmaxf(w2, w3)), w4);
        float rm = __builtin_amdgcn_rcpf(m);
        bool valid = t < len;
        p0 = valid ? w0 * rm : p0;
        p1 = valid ? w1 * rm : p1;
        p2 = valid ? w2 * rm : p2;
        p3 = valid ? w3 * rm : p3;
        p4 = valid ? w4 * rm : p4;
        logacc = valid ? (logacc + __logf(m)) : logacc;
    }

    if (!half && bb < B) {
        lognorm[bb] = __logf(p0 + p1 + p2 + p3 + p4) + logacc;
    }
}

// ---------------------------------------------------------------------------
// Backtrace + unary/binary scores. One thread per batch. bp loads are [s][B]
// -> coalesced; the bp word load is independent of the backtrace dependence
// chain (only the 3-bit field select depends on the running tag).
// ---------------------------------------------------------------------------
__global__ void crf_backtrace_score(const float* __restrict__ pot,
                                    const float* __restrict__ trans,
                                    const int* __restrict__ lens,
                                    const int* __restrict__ tags,
                                    const unsigned short* __restrict__ bp,
                                    const int* __restrict__ last_tag,
                                    const float* __restrict__ lognorm,
                                    float* __restrict__ out, // decoded B*T, then ll B
                                    int B, int T)
{
    __shared__ float str[KTAGS * KTAGS];
    if (threadIdx.x < KTAGS * KTAGS) str[threadIdx.x] = trans[threadIdx.x];
    __syncthreads();

    int b = blockIdx.x * blockDim.x + threadIdx.x;
    if (b >= B) return;

    int len = lens[b];
    int tag = last_tag[b];
    out[(size_t)b * T + (T - 1)] = ((T - 1) < len) ? (float)tag : 0.f;

    int   tg_next = tags[(size_t)b * T + (T - 1)];
    float unary   = ((T - 1) < len)
                  ? pot[((size_t)b * T + (T - 1)) * KTAGS + tg_next] : 0.f;
    float binary  = 0.f;

    for (int s = T - 2; s >= 0; --s) {
        int spf = (s - 8 >= 0) ? (s - 8) : 0;
        __builtin_prefetch(bp + (size_t)spf * B + b, 0, 3);

        unsigned w = bp[(size_t)s * B + b];
        int prev = (int)((w >> (3 * tag)) & 7u);
        out[(size_t)b * T + s] = (s < len) ? (float)prev : 0.f;
        tag = prev;

        int tg = tags[(size_t)b * T + s];
        if ((s + 1) < len) binary += str[tg * KTAGS + tg_next];
        if (s < len)       unary  += pot[((size_t)b * T + s) * KTAGS + tg];
        tg_next = tg;
    }
    out[(size_t)B * T + b] = unary + binary - lognorm[b];
}

// ---------------------------------------------------------------------------
extern "C" void kernel_launch(void* const* d_in, const int* in_sizes, int n_in,
                              void* d_out, int out_size, void* d_ws, size_t ws_size,
                              hipStream_t stream) {
    const float* pot   = (const float*)d_in[0];
    const float* trans = (const float*)d_in[1];
    const int*   lens  = (const int*)d_in[2];
    const int*   tags  = (const int*)d_in[3];
    int B = in_sizes[2];
    int T = in_sizes[3] / B;
    float* out = (float*)d_out;

    // Workspace layout: packed bps [(T-1)][B] u16 | last_tag[B] i32 | lognorm[B] f32
    size_t bpBytes = (size_t)((T > 1) ? (T - 1) : 1) * (size_t)B * sizeof(unsigned short);
    bpBytes = (bpBytes + 255) & ~(size_t)255;
    size_t ltBytes = ((size_t)B * sizeof(int) + 255) & ~(size_t)255;
    unsigned short* bp_ws  = (unsigned short*)d_ws;
    int*            lt_ws  = (int*)((char*)d_ws + bpBytes);
    float*          ln_ws  = (float*)((char*)d_ws + bpBytes + ltBytes);

    const int blk = 256;
    int gv = (B + blk - 1) / blk;            // Viterbi blocks (1 thread/batch)
    int waves = (B + 15) / 16;               // forward: 16 batches per wave32
    int gf = (waves * 32 + blk - 1) / blk;   // forward blocks

    crf_fwd_fused<<<gv + gf, blk, 0, stream>>>(pot, trans, lens,
                                               bp_ws, lt_ws, ln_ws, B, T, gv);

    crf_backtrace_score<<<gv, blk, 0, stream>>>(pot, trans, lens, tags,
                                                bp_ws, lt_ws, ln_ws, out, B, T);
}